// MultiHeadAttentionMyNoattn_46016279610054
// MI455X (gfx1250) — compile-verified
//
#include <hip/hip_runtime.h>
#include <hip/hip_bf16.h>

// Problem dims (fixed by the reference)
#define TGT      256
#define BSZ      8
#define EMB      256
#define SRC_LEN  4096

typedef __attribute__((ext_vector_type(2))) float v2f;
typedef __attribute__((ext_vector_type(8))) float v8f;

// ---------------------------------------------------------------------------
// Kernel 1: per-batch GEMM  C1[b] = mask[b] (256x4096) @ value[:,b,:] (4096x256)
// fp32 WMMA 16x16x4. Block = 256 threads = 8 waves; block computes a 64x64 tile.
// Wave (wm,wn), wm in 0..3, wn in 0..1: rows [wm*16,+16), cols [wn*32,+32)
// as two 16x16 accumulators.
//
// Fragment layout (ISA 7.12.2, 32-bit A 16x4 / B 4x16):
//   lanes 0-15 : v0 = K0, v1 = K1 ;  lanes 16-31 : v0 = K2, v1 = K3
// C/D: VGPR r -> row r (lanes 0-15), row r+8 (lanes 16-31).
//
// LDS layouts:
//   As[64][36]      : row-major, K contiguous -> A fragment = 1x ds_load_b64.
//                     (stride 36 dwords: banks 36*idx mod 64 give 16 distinct
//                      4-dword groups -> both halves conflict-free)
//   BsP[16][160]    : K-paired, element (k,n) at BsP[k>>1][n*2 + (k&1)]
//                     -> B fragment (B[kr][n], B[kr+1][n]) = 1x ds_load_b64,
//                     no register shuffles. Stride 160 dwords (== 32 mod 64)
//                     puts the two wave halves on disjoint bank sets.
// ---------------------------------------------------------------------------
__global__ __launch_bounds__(256) void gemm_mask_value(
    const float* __restrict__ mask,    // (BSZ, TGT, SRC)
    const float* __restrict__ value,   // (SRC, BSZ, EMB)
    float* __restrict__ C1)            // (BSZ, TGT, EMB)
{
    __shared__ float As[64][36];     // 9216 B
    __shared__ float BsP[16][160];   // 10240 B

    const int n0 = blockIdx.x * 64;
    const int m0 = blockIdx.y * 64;
    const int b  = blockIdx.z;

    const int tid  = threadIdx.x;
    const int lane = tid & 31;
    const int wave = tid >> 5;
    const int half = lane >> 4;   // 0: lanes 0-15, 1: lanes 16-31
    const int idx  = lane & 15;
    const int wm   = wave >> 1;   // 0..3
    const int wn   = wave & 1;    // 0..1

    const float* Abase = mask  + ((size_t)b * TGT + m0) * SRC_LEN;
    const float* Bbase = value + (size_t)b * EMB + n0;

    // LDS fill assignments: A tile = 512 float4 (2 per thread), B tile same.
    const int arow = tid >> 3, akq = tid & 7;    // A: rows arow, arow+32; k-quad
    const int bk   = tid >> 4, bnq = tid & 15;   // B: k rows bk, bk+16; n-quad

    v8f acc0 = {};
    v8f acc1 = {};

    for (int kb = 0; kb < SRC_LEN; kb += 32) {
        // ---- stage A (64x32) into LDS, float4 moves ----
        float4 va0 = *(const float4*)(Abase + (size_t)arow        * SRC_LEN + kb + akq * 4);
        float4 va1 = *(const float4*)(Abase + (size_t)(arow + 32) * SRC_LEN + kb + akq * 4);
        *(float4*)&As[arow][akq * 4]      = va0;
        *(float4*)&As[arow + 32][akq * 4] = va1;

        // ---- stage B (32x64) into K-paired LDS layout ----
        float4 vb0 = *(const float4*)(Bbase + (size_t)(kb + bk)      * (BSZ * EMB) + bnq * 4);
        float4 vb1 = *(const float4*)(Bbase + (size_t)(kb + bk + 16) * (BSZ * EMB) + bnq * 4);
        {
            float* d0 = &BsP[bk >> 1][(bk & 1)];          // k = kb+bk
            float* d1 = &BsP[(bk >> 1) + 8][(bk & 1)];    // k = kb+bk+16
            d0[(bnq * 4 + 0) * 2] = vb0.x;
            d0[(bnq * 4 + 1) * 2] = vb0.y;
            d0[(bnq * 4 + 2) * 2] = vb0.z;
            d0[(bnq * 4 + 3) * 2] = vb0.w;
            d1[(bnq * 4 + 0) * 2] = vb1.x;
            d1[(bnq * 4 + 1) * 2] = vb1.y;
            d1[(bnq * 4 + 2) * 2] = vb1.z;
            d1[(bnq * 4 + 3) * 2] = vb1.w;
        }

        // Prefetch next K-tile (emits global_prefetch_b8; uniform condition)
        if (kb + 32 < SRC_LEN) {
            __builtin_prefetch(Abase + (size_t)arow * SRC_LEN + (kb + 32) + akq * 4, 0, 0);
            __builtin_prefetch(Bbase + (size_t)(kb + 32 + bk) * (BSZ * EMB) + bnq * 4, 0, 0);
        }

        __syncthreads();

        // ---- 8 K-steps of 4, 2 WMMAs each; all fragments are b64 LDS loads ----
        #pragma unroll
        for (int kk = 0; kk < 32; kk += 4) {
            const int kr = kk + 2 * half;                 // even
            v2f a  = *(const v2f*)&As[wm * 16 + idx][kr];
            const float* bp = &BsP[kr >> 1][0];
            v2f b0 = *(const v2f*)(bp + (wn * 32 + idx) * 2);
            v2f b1 = *(const v2f*)(bp + (wn * 32 + 16 + idx) * 2);
            acc0 = __builtin_amdgcn_wmma_f32_16x16x4_f32(
                false, a, false, b0, (short)0, acc0, false, false);
            acc1 = __builtin_amdgcn_wmma_f32_16x16x4_f32(
                false, a, false, b1, (short)0, acc1, false, false);
        }

        __syncthreads();
    }

    // ---- epilogue: write 16x32 wave strip to C1 (b, t, e) ----
    float* crow = C1 + ((size_t)b * TGT + m0 + wm * 16 + 8 * half) * EMB
                     + n0 + wn * 32 + idx;
    #pragma unroll
    for (int r = 0; r < 8; ++r) {
        crow[(size_t)r * EMB]      = acc0[r];
        crow[(size_t)r * EMB + 16] = acc1[r];
    }
}

// ---------------------------------------------------------------------------
// Kernel 2: out = C1 (2048x256) @ W^T (256x256) + bias, scattered to
// (tgt, bsz, E). One 16x16 WMMA tile per wave; B fragment rows are W rows
// (B[k][n] = W[n][k]) so both A and B fragments are contiguous v2f loads.
// Grid: 256 blocks x 8 waves = 2048 tiles = 128 m-tiles x 16 n-tiles.
// ---------------------------------------------------------------------------
__global__ __launch_bounds__(256) void proj_kernel(
    const float* __restrict__ C1,     // (BSZ*TGT, EMB), row = b*TGT + t
    const float* __restrict__ W,      // (EMB, EMB) row-major
    const float* __restrict__ bias,   // (EMB)
    float* __restrict__ out)          // (TGT, BSZ, EMB)
{
    const int tid  = threadIdx.x;
    const int lane = tid & 31;
    const int wave = tid >> 5;
    const int half = lane >> 4;
    const int idx  = lane & 15;

    const int tile = blockIdx.x * 8 + wave;
    const int m0   = (tile >> 4) * 16;
    const int n0   = (tile & 15) * 16;

    v8f acc = {};
    const float* arow = C1 + (size_t)(m0 + idx) * EMB + 2 * half;
    const float* brow = W  + (size_t)(n0 + idx) * EMB + 2 * half;

    #pragma unroll 8
    for (int kb = 0; kb < EMB; kb += 4) {
        v2f a  = *(const v2f*)(arow + kb);
        v2f bf = *(const v2f*)(brow + kb);
        acc = __builtin_amdgcn_wmma_f32_16x16x4_f32(
            false, a, false, bf, (short)0, acc, false, false);
    }

    const float bv = bias[n0 + idx];
    #pragma unroll
    for (int r = 0; r < 8; ++r) {
        const int g  = m0 + r + 8 * half;  // row in (b*TGT + t)
        const int bb = g >> 8;             // batch
        const int tt = g & 255;            // target position
        out[((size_t)tt * BSZ + bb) * EMB + n0 + idx] = acc[r] + bv;
    }
}

// ---------------------------------------------------------------------------
// Kernel 3: attn_weights = proposal_mask (exact copy), float4 grid-stride.
// ---------------------------------------------------------------------------
__global__ __launch_bounds__(256) void copy_mask_kernel(
    const float4* __restrict__ src, float4* __restrict__ dst, int n4)
{
    int i      = blockIdx.x * blockDim.x + threadIdx.x;
    int stride = gridDim.x * blockDim.x;
    for (; i < n4; i += stride) dst[i] = src[i];
}

// ---------------------------------------------------------------------------
extern "C" void kernel_launch(void* const* d_in, const int* in_sizes, int n_in,
                              void* d_out, int out_size, void* d_ws, size_t ws_size,
                              hipStream_t stream)
{
    (void)in_sizes; (void)n_in; (void)out_size; (void)ws_size;

    // setup_inputs order: query, key, value, proposal_mask, out_proj_weight, out_proj_bias
    const float* value = (const float*)d_in[2];
    const float* mask  = (const float*)d_in[3];
    const float* W     = (const float*)d_in[4];
    const float* bias  = (const float*)d_in[5];
    // query (d_in[0]) and key (d_in[1]) are shape-only in the reference.

    float* out      = (float*)d_out;
    float* attn_out = out;                                  // TGT*BSZ*EMB floats
    float* attn_w   = out + (size_t)TGT * BSZ * EMB;        // BSZ*TGT*SRC floats
    float* C1       = (float*)d_ws;                         // BSZ*TGT*EMB floats (2 MB)

    // 1) C1[b] = mask[b] @ value[:,b,:]
    gemm_mask_value<<<dim3(4, 4, BSZ), 256, 0, stream>>>(mask, value, C1);

    // 2) attn_out = C1 @ W^T + bias (scattered to (tgt, bsz, E))
    proj_kernel<<<256, 256, 0, stream>>>(C1, W, bias, attn_out);

    // 3) attn_weights = proposal_mask
    const int n4 = (BSZ * TGT * SRC_LEN) / 4;               // 2,097,152 float4
    copy_mask_kernel<<<2048, 256, 0, stream>>>(
        (const float4*)mask, (float4*)attn_w, n4);
}